// RBM_36455682408785
// MI455X (gfx1250) — compile-verified
//
#include <hip/hip_runtime.h>
#include <stdint.h>

// Ising-chain Metropolis-Hastings: one workgroup (8 wave32s) per chain.
// Spins + couplings live in LDS for all 64 steps; HBM touched only for
// initial read (async DMA to LDS) and final write (async DMA from LDS).

static constexpr int NSPIN = 2048;   // spins per chain
static constexpr int TPB   = 256;    // threads per block (8 waves)
static constexpr int PER_T = NSPIN / TPB;   // 8 spins per thread
static constexpr int NWAVE = TPB / 32;      // 8
static constexpr int NCW   = NSPIN / 32;    // 64 packed change-bit words

// Stateless counter-based hash (statistical stand-in for threefry).
__device__ __forceinline__ uint32_t hash32(uint32_t a, uint32_t b, uint32_t c) {
  uint32_t h = a * 0x9E3779B1u + b * 0x85EBCA77u + c * 0xC2B2AE3Du + 0x27D4EB2Fu;
  h ^= h >> 16; h *= 0x7FEB352Du;
  h ^= h >> 15; h *= 0x846CA68Bu;
  h ^= h >> 16;
  return h;
}

// Generic-space shared pointers carry the workgroup-relative LDS byte offset in
// their low 32 bits (flat->LDS mapping truncates; HW adds LDS_BASE).
__device__ __forceinline__ uint32_t lds_off_u32(const void* p) {
  return (uint32_t)(uintptr_t)p;
}

// CDNA5 async DMA: 16 bytes/lane, tracked with ASYNCcnt.
__device__ __forceinline__ void async_load_b128(uint32_t lds_byte_off, const void* g) {
  asm volatile("global_load_async_to_lds_b128 %0, %1, off"
               :: "v"(lds_byte_off), "v"((uint64_t)(uintptr_t)g)
               : "memory");
}
__device__ __forceinline__ void async_store_b128(const void* g, uint32_t lds_byte_off) {
  asm volatile("global_store_async_from_lds_b128 %0, %1, off"
               :: "v"((uint64_t)(uintptr_t)g), "v"(lds_byte_off)
               : "memory");
}
__device__ __forceinline__ void wait_async0() {
  asm volatile("s_wait_asynccnt 0" ::: "memory");
}

__global__ __launch_bounds__(TPB)
void ising_mh_kernel(const float* __restrict__ v, const float* __restrict__ J,
                     const int* __restrict__ kptr, float* __restrict__ out, int B) {
  __shared__ __align__(16) float sJ[NSPIN];   // couplings          (8 KB)
  __shared__ __align__(16) float sS[NSPIN];   // spin row, +-1      (8 KB)
  __shared__ uint32_t sC[NCW];                // packed change bits
  __shared__ float    sRed[NWAVE];
  __shared__ int      sAcc;

  const int t    = threadIdx.x;
  const int b    = blockIdx.x;
  const int lane = t & 31;
  const int wave = t >> 5;

  // ---- stage J and this chain's v row into LDS via async DMA ----
  {
    const char* gJ = (const char*)J;
    const char* gV = (const char*)(v + (size_t)b * NSPIN);
    const uint32_t oJ = lds_off_u32(sJ) + (uint32_t)t * 16u;
    const uint32_t oS = lds_off_u32(sS) + (uint32_t)t * 16u;
    async_load_b128(oJ,                  gJ + t * 16);
    async_load_b128(oJ + TPB * 16u,      gJ + TPB * 16 + t * 16);
    async_load_b128(oS,                  gV + t * 16);
    async_load_b128(oS + TPB * 16u,      gV + TPB * 16 + t * 16);
    wait_async0();
  }
  __syncthreads();

  // ---- map {0,1} -> {-1,+1} in LDS; emit first output (2v-1) ----
  float* out0 = out + (size_t)b * NSPIN;
  #pragma unroll
  for (int i = 0; i < PER_T; ++i) {
    const int n = i * TPB + t;          // strided: conflict-free DS access
    const float s = 2.0f * sS[n] - 1.0f;
    sS[n]   = s;
    out0[n] = s;                        // coalesced global store
  }
  __syncthreads();

  const int K = *kptr;
  for (int k = 0; k < K; ++k) {
    // 64 hash words give all 2048 per-spin flip proposals for this (k, b).
    if (t < NCW) sC[t] = hash32(0xC0FFEEu ^ (uint32_t)k, (uint32_t)b, (uint32_t)t);
    __syncthreads();

    // neg_energy_diff = sum over bonds with c[n] != c[n+1] of 2*J*s[n]*s[n+1]
    float partial = 0.0f;
    #pragma unroll
    for (int i = 0; i < PER_T; ++i) {
      const int n  = i * TPB + t;
      const int np = (n + 1) & (NSPIN - 1);
      const uint32_t bn  = (sC[n  >> 5] >> (n  & 31)) & 1u;
      const uint32_t bnp = (sC[np >> 5] >> (np & 31)) & 1u;
      if (bn != bnp)
        partial += 2.0f * sJ[n] * sS[n] * sS[np];
    }
    // wave32 butterfly reduce, then 8 wave partials through LDS
    #pragma unroll
    for (int off = 16; off > 0; off >>= 1)
      partial += __shfl_xor(partial, off, 32);
    if (lane == 0) sRed[wave] = partial;
    __syncthreads();

    if (t == 0) {
      float diff = 0.0f;
      #pragma unroll
      for (int w = 0; w < NWAVE; ++w) diff += sRed[w];
      const float p  = __expf(-fmaxf(diff, 0.0f));        // min(1, exp(-dE))
      const uint32_t hu = hash32(0x00ACCE9Du ^ (uint32_t)k, (uint32_t)b, 0x5EEDu);
      const float u  = (float)(hu >> 8) * (1.0f / 16777216.0f);
      sAcc = (u < p) ? 1 : 0;
    }
    __syncthreads();

    if (sAcc) {
      #pragma unroll
      for (int i = 0; i < PER_T; ++i) {
        const int n = i * TPB + t;
        if ((sC[n >> 5] >> (n & 31)) & 1u) sS[n] = -sS[n];
      }
    }
    __syncthreads();
  }

  // ---- drain final state: async LDS -> global ----
  {
    const char* gO = (const char*)(out + (size_t)B * NSPIN + (size_t)b * NSPIN);
    const uint32_t oS = lds_off_u32(sS) + (uint32_t)t * 16u;
    async_store_b128(gO + t * 16,            oS);
    async_store_b128(gO + TPB * 16 + t * 16, oS + TPB * 16u);
    wait_async0();
  }
}

extern "C" void kernel_launch(void* const* d_in, const int* in_sizes, int n_in,
                              void* d_out, int out_size, void* d_ws, size_t ws_size,
                              hipStream_t stream) {
  const float* v  = (const float*)d_in[0];   // [B, N] spins in {0,1}
  const float* J  = (const float*)d_in[1];   // [N]   couplings
  const int*   kp = (const int*)d_in[2];     // scalar step count (device mem)
  float*       o  = (float*)d_out;           // [2, B, N]: (2v-1, vn)
  const int B = in_sizes[0] / NSPIN;
  ising_mh_kernel<<<B, TPB, 0, stream>>>(v, J, kp, o, B);
}